// TransferFunctionModel_61108794687953
// MI455X (gfx1250) — compile-verified
//
#include <hip/hip_runtime.h>
#include <hip/hip_bf16.h>

// ---------------------------------------------------------------------------
// Problem constants (from the reference)
// ---------------------------------------------------------------------------
#define BATCH      4
#define CHN        128
#define NSAMP      32768
#define NATOMS     512
#define ATOMLEN    512
#define TRLEN      16384
#define NEV        16
#define NPAIR      (BATCH * NEV)       // 64
#define LRELU_K    0.2f
#define CONV_R     16                  // outputs per thread in conv kernel

typedef __attribute__((ext_vector_type(2))) float v2f;
typedef __attribute__((ext_vector_type(8))) float v8f;

// ---------------------------------------------------------------------------
// K1: attention scores  attn[b,t] = relu( sum_c x[b,c,t]*w[c] + bias )
// Each thread produces 4 consecutive t via float4 (coalesced b128 on the
// 64MB x stream); w[c] is wave-uniform -> scalar loads.
// ---------------------------------------------------------------------------
__global__ void attn_kernel(const float* __restrict__ x,
                            const float* __restrict__ w,
                            const float* __restrict__ bias,
                            float* __restrict__ attn) {
  int gid = blockIdx.x * blockDim.x + threadIdx.x;      // over B*NSAMP/4
  if (gid >= BATCH * (NSAMP / 4)) return;
  int b  = gid / (NSAMP / 4);
  int t4 = gid - b * (NSAMP / 4);
  const float4* xb = (const float4*)(x + (size_t)b * CHN * NSAMP) + t4;
  float4 acc; acc.x = acc.y = acc.z = acc.w = bias[0];
  #pragma unroll 8
  for (int c = 0; c < CHN; ++c) {
    float4 xv = xb[(size_t)c * (NSAMP / 4)];
    float wc = w[c];
    acc.x += xv.x * wc; acc.y += xv.y * wc;
    acc.z += xv.z * wc; acc.w += xv.w * wc;
  }
  acc.x = acc.x > 0.0f ? acc.x : 0.0f;
  acc.y = acc.y > 0.0f ? acc.y : 0.0f;
  acc.z = acc.z > 0.0f ? acc.z : 0.0f;
  acc.w = acc.w > 0.0f ? acc.w : 0.0f;
  ((float4*)attn)[gid] = acc;
}

// ---------------------------------------------------------------------------
// K2: top-16 per batch.  One block (1024 threads) per batch; the 32768-elem
// score row lives in LDS (dynamic), 16 masked argmax passes.
// Tie-break: higher value wins; equal value -> lower index (matches lax.top_k).
// ---------------------------------------------------------------------------
__global__ void topk_kernel(const float* __restrict__ attn,
                            float* __restrict__ vals,
                            int* __restrict__ idxs) {
  extern __shared__ float smem[];
  float* sa = smem;                       // 32768 floats
  float* rv = smem + NSAMP;               // 1024 floats
  int*   ri = (int*)(rv + 1024);          // 1024 ints
  const int tid = threadIdx.x;
  const int b   = blockIdx.x;

  for (int t = tid; t < NSAMP; t += 1024)
    sa[t] = attn[b * NSAMP + t];
  __syncthreads();

  for (int pass = 0; pass < NEV; ++pass) {
    float bv = -1e30f; int bi = 0x7fffffff;
    for (int t = tid; t < NSAMP; t += 1024) {
      float v = sa[t];
      if (v > bv || (v == bv && t < bi)) { bv = v; bi = t; }
    }
    rv[tid] = bv; ri[tid] = bi;
    __syncthreads();
    for (int s = 512; s > 0; s >>= 1) {
      if (tid < s) {
        float v2 = rv[tid + s]; int i2 = ri[tid + s];
        if (v2 > rv[tid] || (v2 == rv[tid] && i2 < ri[tid])) {
          rv[tid] = v2; ri[tid] = i2;
        }
      }
      __syncthreads();
    }
    if (tid == 0) {
      vals[b * NEV + pass] = rv[0];
      idxs[b * NEV + pass] = ri[0];
      sa[ri[0]] = -1e30f;               // mask winner
    }
    __syncthreads();
  }
}

// ---------------------------------------------------------------------------
// K3: lat[p, c] = x[b, c, idx[p]] * vals[p]   (p = b*16+e, 64x128)
// ---------------------------------------------------------------------------
__global__ void lat_kernel(const float* __restrict__ x,
                           const float* __restrict__ vals,
                           const int* __restrict__ idxs,
                           float* __restrict__ lat) {
  int p = blockIdx.x;           // 0..63
  int c = threadIdx.x;          // 0..127
  int b = p >> 4;
  int t = idxs[p];
  lat[p * CHN + c] = x[(size_t)b * CHN * NSAMP + (size_t)c * NSAMP + t] * vals[p];
}

// ---------------------------------------------------------------------------
// K4: fp32 WMMA GEMM  C(64xN) = act(A(64xK) * B(KxN) + bias)
// Templated on K,N so strides are compile-time (immediate-offset load
// clauses, no 64-bit address muls); K-loop fully unrolled so the scheduler
// interleaves loads across V_WMMA_F32_16X16X4_F32 issues.
// Fragment layout per CDNA5 ISA 7.12.2:
//   A 16x4 : lane = m (2 halves), vgpr j -> K = kbase + 2*(lane>>4) + j
//   B 4x16 : lane = n (2 halves), same K mapping
//   C 16x16: vgpr r, lane l -> M = (l>>4)*8 + r, N = l&15
// ---------------------------------------------------------------------------
template <int K, int N>
__global__ void gemm_wmma_f32(const float* __restrict__ A,
                              const float* __restrict__ B,
                              const float* __restrict__ bias,
                              float* __restrict__ C,
                              int act, int totalTiles) {
  const int wave = threadIdx.x >> 5;
  const int lane = threadIdx.x & 31;
  const int tile = blockIdx.x * (blockDim.x >> 5) + wave;
  if (tile >= totalTiles) return;

  const int tm = tile & 3;                // M = 64 -> 4 M-tiles
  const int tn = tile >> 2;

  const int mfrag = lane & 15;
  const int kh    = lane >> 4;            // K-half selector
  const float* Arow = A + (size_t)(tm * 16 + mfrag) * K + 2 * kh;
  const int    col  = tn * 16 + mfrag;    // N column for B/C fragments
  const float* Bcol = B + col + (size_t)2 * kh * N;

  v8f c = {};
  #pragma unroll
  for (int k0 = 0; k0 < K; k0 += 4) {
    if ((k0 & 31) == 0 && k0 + 32 < K)     // speculative -> global_prefetch_b8
      __builtin_prefetch(&Bcol[(size_t)(k0 + 32) * N], 0, 1);
    v2f a = *(const v2f*)(Arow + k0);      // 8B-aligned: global_load_b64
    v2f bfr;
    bfr.x = Bcol[(size_t)k0 * N];
    bfr.y = Bcol[(size_t)(k0 + 1) * N];
    c = __builtin_amdgcn_wmma_f32_16x16x4_f32(
            /*neg_a=*/false, a, /*neg_b=*/false, bfr,
            /*c_mod=*/(short)0, c, /*reuse_a=*/false, /*reuse_b=*/false);
  }

  float bv = bias[col];
  #pragma unroll
  for (int r = 0; r < 8; ++r) {
    int mrow = tm * 16 + kh * 8 + r;
    float v = c[r] + bv;
    if (act) v = v > 0.0f ? v : LRELU_K * v;
    C[(size_t)mrow * N + col] = v;
  }
}

// ---------------------------------------------------------------------------
// K5: per-row argmax over N (hard_softmax forward == one_hot(argmax))
// ---------------------------------------------------------------------------
__global__ void argmax_kernel(const float* __restrict__ logits, int N,
                              int* __restrict__ outIdx) {
  __shared__ float rv[256];
  __shared__ int   ri[256];
  const int row = blockIdx.x;
  const int tid = threadIdx.x;
  const float* L = logits + (size_t)row * N;
  float bv = -1e30f; int bi = 0x7fffffff;
  for (int t = tid; t < N; t += 256) {
    float v = L[t];
    if (v > bv || (v == bv && t < bi)) { bv = v; bi = t; }
  }
  rv[tid] = bv; ri[tid] = bi;
  __syncthreads();
  for (int s = 128; s > 0; s >>= 1) {
    if (tid < s) {
      float v2 = rv[tid + s]; int i2 = ri[tid + s];
      if (v2 > rv[tid] || (v2 == rv[tid] && i2 < ri[tid])) {
        rv[tid] = v2; ri[tid] = i2;
      }
    }
    __syncthreads();
  }
  if (tid == 0) outIdx[row] = ri[0];
}

// ---------------------------------------------------------------------------
// K6: direct causal conv per (b,ev) pair.  atoms_p occupies only [0,512), so
// fft_convolve truncated to TRLEN == ev[n] = sum_{k<=min(511,n)} a[k]*t[n-k],
// plus the atoms_p residual.  Rows staged in LDS (2KB + 64KB) with float4.
// Register sliding window: CONV_R consecutive outputs per thread; per k-step
// -> CONV_R FMAs, 1 broadcast a-read, 1 new t-read (window shift).  Guarded
// shift loads zero-fill t[n-k] for n<k, reproducing the causal zero-padding.
// ---------------------------------------------------------------------------
__global__ void conv_kernel(const float* __restrict__ atoms_dict,
                            const float* __restrict__ transfer_dict,
                            const int* __restrict__ ia,
                            const int* __restrict__ it,
                            float* __restrict__ ev) {
  extern __shared__ float smem[];
  float* a_s = smem;             // 512
  float* t_s = smem + ATOMLEN;   // 16384
  const int p   = blockIdx.x;
  const int tid = threadIdx.x;   // 256 threads
  const float4* arow4 = (const float4*)(atoms_dict    + (size_t)ia[p] * ATOMLEN);
  const float4* trow4 = (const float4*)(transfer_dict + (size_t)it[p] * TRLEN);
  for (int i = tid; i < ATOMLEN / 4; i += 256) ((float4*)a_s)[i] = arow4[i];
  for (int i = tid; i < TRLEN / 4;   i += 256) ((float4*)t_s)[i] = trow4[i];
  __syncthreads();

  float* evp = ev + (size_t)p * TRLEN;
  for (int n0 = 0; n0 < TRLEN; n0 += 256 * CONV_R) {   // 4 passes
    const int nlo = n0 + tid * CONV_R;
    float acc[CONV_R];
    float w[CONV_R];
    #pragma unroll
    for (int j = 0; j < CONV_R; ++j) { acc[j] = 0.0f; w[j] = t_s[nlo + j]; }

    #pragma unroll 8
    for (int k = 0; k < ATOMLEN; ++k) {
      float av = a_s[k];                       // uniform addr -> LDS broadcast
      #pragma unroll
      for (int j = 0; j < CONV_R; ++j) acc[j] += av * w[j];
      int src = nlo - (k + 1);                 // slide window for next k
      #pragma unroll
      for (int j = CONV_R - 1; j >= 1; --j) w[j] = w[j - 1];
      w[0] = (src >= 0) ? t_s[src] : 0.0f;     // zero-fill = causal padding
    }

    #pragma unroll
    for (int j = 0; j < CONV_R; ++j) {
      int n = nlo + j;
      float v = acc[j];
      if (n < ATOMLEN) v += a_s[n];            // + atoms_p residual
      evp[n] = v;
    }
  }
}

// ---------------------------------------------------------------------------
// K7: deterministic scatter-as-gather. Reference scatter-adds all 64 ev rows
// at offsets idx[p] into one shared buffer, broadcast over batch, keep [:N].
// out[b*N + t] = sum_p ev[p][t - idx[p]]  (fixed p order -> replay-stable,
// no float atomics).
// ---------------------------------------------------------------------------
__global__ void out_kernel(const float* __restrict__ ev,
                           const int* __restrict__ idxs,
                           float* __restrict__ out) {
  __shared__ int s_idx[NPAIR];
  if (threadIdx.x < NPAIR) s_idx[threadIdx.x] = idxs[threadIdx.x];
  __syncthreads();
  int t = blockIdx.x * blockDim.x + threadIdx.x;   // 0..NSAMP-1
  if (t >= NSAMP) return;
  float acc = 0.0f;
  for (int p = 0; p < NPAIR; ++p) {
    int d = t - s_idx[p];
    if (d >= 0 && d < TRLEN) acc += ev[(size_t)p * TRLEN + d];
  }
  #pragma unroll
  for (int b = 0; b < BATCH; ++b)
    out[(size_t)b * NSAMP + t] = acc;
}

// ---------------------------------------------------------------------------
// Launch
// ---------------------------------------------------------------------------
extern "C" void kernel_launch(void* const* d_in, const int* in_sizes, int n_in,
                              void* d_out, int out_size, void* d_ws, size_t ws_size,
                              hipStream_t stream) {
  const float* x       = (const float*)d_in[0];
  const float* attn_w  = (const float*)d_in[1];
  const float* attn_b  = (const float*)d_in[2];
  const float* wa1     = (const float*)d_in[3];
  const float* ba1     = (const float*)d_in[4];
  const float* wa2     = (const float*)d_in[5];
  const float* ba2     = (const float*)d_in[6];
  const float* wa3     = (const float*)d_in[7];
  const float* ba3     = (const float*)d_in[8];
  const float* wt1     = (const float*)d_in[9];
  const float* bt1     = (const float*)d_in[10];
  const float* wt2     = (const float*)d_in[11];
  const float* bt2     = (const float*)d_in[12];
  const float* wt3     = (const float*)d_in[13];
  const float* bt3     = (const float*)d_in[14];
  const float* atoms_d = (const float*)d_in[15];
  const float* trans_d = (const float*)d_in[16];
  float* out = (float*)d_out;

  // Workspace layout (floats). Total ~4.95 MB.
  float* ws       = (float*)d_ws;
  float* attn     = ws;                           // 131072
  float* lat      = attn + BATCH * NSAMP;         // 8192
  float* h1       = lat  + NPAIR * CHN;           // 8192
  float* h2       = h1   + NPAIR * CHN;           // 8192
  float* logits_a = h2   + NPAIR * CHN;           // 32768
  float* logits_t = logits_a + NPAIR * NATOMS;    // 1048576 (reused as ev)
  float* vals     = logits_t + NPAIR * TRLEN;     // 64
  int*   idxs     = (int*)(vals + NPAIR);         // 64
  int*   ia       = idxs + NPAIR;                 // 64
  int*   it       = ia + NPAIR;                   // 64
  float* ev       = logits_t;                     // reuse after argmax(it)

  // 1) attention scores (float4 per thread)
  attn_kernel<<<(BATCH * (NSAMP / 4) + 255) / 256, 256, 0, stream>>>(x, attn_w, attn_b, attn);

  // 2) top-16 per batch (dynamic LDS: row + reduction arrays)
  size_t topk_lds = (NSAMP + 1024 + 1024) * sizeof(float);
  topk_kernel<<<BATCH, 1024, topk_lds, stream>>>(attn, vals, idxs);

  // 3) gather + scale latents
  lat_kernel<<<NPAIR, CHN, 0, stream>>>(x, vals, idxs, lat);

  // 4) atoms MLP: 64x128x128 (x2, lrelu) then 64x128x512, argmax
  int tiles_cc = (NPAIR / 16) * (CHN / 16);       // 32 tiles
  gemm_wmma_f32<CHN, CHN><<<(tiles_cc + 3) / 4, 128, 0, stream>>>(lat, wa1, ba1, h1, 1, tiles_cc);
  gemm_wmma_f32<CHN, CHN><<<(tiles_cc + 3) / 4, 128, 0, stream>>>(h1,  wa2, ba2, h2, 1, tiles_cc);
  int tiles_a = (NPAIR / 16) * (NATOMS / 16);     // 128 tiles
  gemm_wmma_f32<CHN, NATOMS><<<(tiles_a + 3) / 4, 128, 0, stream>>>(h2, wa3, ba3, logits_a, 0, tiles_a);
  argmax_kernel<<<NPAIR, 256, 0, stream>>>(logits_a, NATOMS, ia);

  // 5) transfers MLP: same trunk shapes, head 64x128x16384, argmax
  gemm_wmma_f32<CHN, CHN><<<(tiles_cc + 3) / 4, 128, 0, stream>>>(lat, wt1, bt1, h1, 1, tiles_cc);
  gemm_wmma_f32<CHN, CHN><<<(tiles_cc + 3) / 4, 128, 0, stream>>>(h1,  wt2, bt2, h2, 1, tiles_cc);
  int tiles_t = (NPAIR / 16) * (TRLEN / 16);      // 4096 tiles
  gemm_wmma_f32<CHN, TRLEN><<<(tiles_t + 3) / 4, 128, 0, stream>>>(h2, wt3, bt3, logits_t, 0, tiles_t);
  argmax_kernel<<<NPAIR, 256, 0, stream>>>(logits_t, TRLEN, it);

  // 6) direct conv (dict rows staged in LDS), writes ev (reuses logits_t)
  size_t conv_lds = (ATOMLEN + TRLEN) * sizeof(float);
  conv_kernel<<<NPAIR, 256, conv_lds, stream>>>(atoms_d, trans_d, ia, it, ev);

  // 7) deterministic scatter-gather + batch broadcast
  out_kernel<<<(NSAMP + 255) / 256, 256, 0, stream>>>(ev, idxs, out);
}